// Piglet_82497731821642
// MI455X (gfx1250) — compile-verified
//
#include <hip/hip_runtime.h>

// ---------------------------------------------------------------------------
// PIGLET signed-graph GNN forward for MI455X (gfx1250, wave32).
//
//  * Dense node GEMMs (x@W, z@W, final z@Wout) use V_WMMA_F32_16X16X4_F32
//    (full fp32 precision, 16x16 tile per wave, K stepped by 4).
//  * Edge attention: tanh bounds the logit to (-1,1), so exp() cannot
//    overflow and the segment-max pass of the softmax is dropped
//    (exp(e-m)/sum == exp(e)/sum exactly). Each propagate = 2 edge passes.
//  * Wave-per-edge: lane k owns feature k -> coalesced 128B row gathers,
//    shfl_xor butterfly for the 64-wide dot product, one
//    global_atomic_add_f32 instruction scatters the whole 32-float message.
//
// Input order assumption: setup_inputs() dict insertion order, with the
// nested 'params' dict expanded in ITS insertion order:
//   0 x | 1..26 params (c1_Wb,c1_bb,c1_Wu,c1_bu,c1_Wb2b,c1_bb2b,c1_Wu2b,
//   c1_bu2b,c1_Wu2u,c1_bu2u,c1_Wb2u,c1_bb2u,c2_Wb,c2_bb,c2_Wu,c2_bu,
//   c2_Wb2b,c2_bb2b,c2_Wu2b,c2_bu2b,c2_Wu2u,c2_bu2u,c2_Wb2u,c2_bb2u,
//   Wout,bout) | 27 pos_edge_index | 28 neg_edge_index | 29 pos_weight |
//   30 neg_weight
//
// Workspace: 3*N*64 + N + (Ep+N+En) floats ~= 48.6 MB.
// ---------------------------------------------------------------------------

typedef float v2f __attribute__((ext_vector_type(2)));
typedef float v8f __attribute__((ext_vector_type(8)));

// C[m0:m0+16, n0:n0+16] = A[:, acol:acol+K] @ [Wl | Wr] + [bl | br]
// A row stride lda, W row stride ldw*, C row stride ldc. 4 waves/block cover
// 64 output columns; gridDim.x = Nrows/16 (Nrows divisible by 16).
__global__ __launch_bounds__(128) void gemm16_wmma(
    const float* __restrict__ A, int lda, int acol, int K,
    const float* __restrict__ Wl, int ldwl,
    const float* __restrict__ Wr, int ldwr,
    const float* __restrict__ bl, const float* __restrict__ br,
    float* __restrict__ C, int ldc, int doTanh)
{
  const int lane = threadIdx.x & 31;
  const int wv   = threadIdx.x >> 5;          // 0..3 -> output column tile
  const int l    = lane & 15;
  const int h    = lane >> 4;                 // 0 or 1 (half-wave)
  const int m0   = blockIdx.x << 4;
  const int n0   = wv << 4;

  const float* W   = (n0 < 32) ? Wl : Wr;
  const int    ldw = (n0 < 32) ? ldwl : ldwr;
  const int    nc  = (n0 < 32) ? n0 : (n0 - 32);
  const float  bv  = ((n0 < 32) ? bl : br)[nc + l];

  v8f acc = {0.f, 0.f, 0.f, 0.f, 0.f, 0.f, 0.f, 0.f};
  const float* Arow = A + (m0 + l) * lda + acol;

  for (int k0 = 0; k0 < K; k0 += 4) {
    // A 16x4 fp32 layout: lanes 0-15 hold K=k0,k0+1 ; lanes 16-31 K=k0+2,k0+3
    v2f a, b;
    a.x = Arow[k0 + 2 * h];
    a.y = Arow[k0 + 2 * h + 1];
    // B 4x16 fp32 layout: VGPR r, lanes 0-15 -> K=r ; lanes 16-31 -> K=r+2
    b.x = W[(k0 + 2 * h)     * ldw + nc + l];
    b.y = W[(k0 + 2 * h + 1) * ldw + nc + l];
    acc = __builtin_amdgcn_wmma_f32_16x16x4_f32(
        false, a, false, b, (short)0, acc, false, false);
  }
  // C/D 16x16 fp32 layout: VGPR r -> row m0 + r + 8*h, col n0 + l
  for (int r = 0; r < 8; ++r) {
    float v = acc[r] + bv;
    if (doTanh) v = tanhf(v);
    C[(m0 + r + 8 * h) * ldc + n0 + l] = v;
  }
}

// Edge id decode (shared by both passes):
//   e <  EA        : segment A edge (p=0)
//   e <  EA+SA     : self-loop, src=dst=e-EA, w=1 (p=0)
//   else           : segment B edge (p=1)
__device__ __forceinline__ void decode_edge(
    int e, const int* __restrict__ srcA, const int* __restrict__ dstA,
    int EA, int SA, const int* __restrict__ srcB, const int* __restrict__ dstB,
    int& src, int& dst, int& p)
{
  if (e < EA)            { src = srcA[e]; dst = dstA[e]; p = 0; }
  else if (e < EA + SA)  { src = dst = e - EA;           p = 0; }
  else { int j = e - EA - SA; src = srcB[j]; dst = dstB[j]; p = 1; }
}

// Pass 1: per edge compute ex = exp(tanh([xj,xi] . W + b)); escr[e]=ex;
// s[dst] += ex.  One wave32 per edge, lane k owns feature k.
__global__ __launch_bounds__(256) void edge_pass1(
    const int* __restrict__ srcA, const int* __restrict__ dstA, int EA, int SA,
    const int* __restrict__ srcB, const int* __restrict__ dstB, int EB,
    const float* __restrict__ x1, const float* __restrict__ x2, int xs,
    const float* __restrict__ W0, const float* __restrict__ b0p,
    const float* __restrict__ W1, const float* __restrict__ b1p,
    float* __restrict__ escr, float* __restrict__ s)
{
  const int ET   = EA + SA + EB;
  const int lane = threadIdx.x & 31;
  const int wid  = (int)((blockIdx.x * blockDim.x + threadIdx.x) >> 5);
  const int nw   = (int)((gridDim.x * blockDim.x) >> 5);
  const float bias0 = b0p[0];
  const float bias1 = b1p[0];

  for (int e = wid; e < ET; e += nw) {
    int src, dst, p;
    decode_edge(e, srcA, dstA, EA, SA, srcB, dstB, src, dst, p);
    const float* x = p ? x2 : x1;
    const float* W = p ? W1 : W0;
    float xj = x[src * xs + lane];              // coalesced 128B row
    float xi = x[dst * xs + lane];
    float partial = xj * W[lane] + xi * W[32 + lane];
    // wave32 butterfly sum
    for (int off = 16; off; off >>= 1)
      partial += __shfl_xor(partial, off, 32);
    float ex = expf(tanhf(partial + (p ? bias1 : bias0)));
    if (lane == 0) {
      escr[e] = ex;
      atomicAdd(&s[dst], ex);
    }
  }
}

// Pass 2: coef = escr[e]/(s[dst]+1e-16)*w; acc[dst, k] += xj[k]*coef.
// One wave per edge -> a single 32-lane global_atomic_add_f32 per edge.
__global__ __launch_bounds__(256) void edge_pass2(
    const int* __restrict__ srcA, const int* __restrict__ dstA,
    const float* __restrict__ wA, int EA, int SA,
    const int* __restrict__ srcB, const int* __restrict__ dstB,
    const float* __restrict__ wB, int EB,
    const float* __restrict__ x1, const float* __restrict__ x2, int xs,
    const float* __restrict__ escr, const float* __restrict__ s,
    float* __restrict__ acc, int accs, int accOff)
{
  const int ET   = EA + SA + EB;
  const int lane = threadIdx.x & 31;
  const int wid  = (int)((blockIdx.x * blockDim.x + threadIdx.x) >> 5);
  const int nw   = (int)((gridDim.x * blockDim.x) >> 5);

  for (int e = wid; e < ET; e += nw) {
    int src, dst, p;
    decode_edge(e, srcA, dstA, EA, SA, srcB, dstB, src, dst, p);
    float wgt = (e < EA) ? wA[e] : ((e < EA + SA) ? 1.0f : wB[e - EA - SA]);
    float coef = escr[e] / (s[dst] + 1e-16f) * wgt;   // broadcast loads
    const float* x = p ? x2 : x1;
    float xj = x[src * xs + lane];
    atomicAdd(&acc[dst * accs + accOff + lane], xj * coef);
  }
}

__global__ __launch_bounds__(256) void fill_f32(float* __restrict__ p,
                                                int n, float v) {
  for (int i = blockIdx.x * blockDim.x + threadIdx.x; i < n;
       i += gridDim.x * blockDim.x)
    p[i] = v;
}

__global__ __launch_bounds__(256) void tanh_ip(float* __restrict__ p, int n) {
  for (int i = blockIdx.x * blockDim.x + threadIdx.x; i < n;
       i += gridDim.x * blockDim.x)
    p[i] = tanhf(p[i]);
}

extern "C" void kernel_launch(void* const* d_in, const int* in_sizes, int n_in,
                              void* d_out, int out_size, void* d_ws,
                              size_t ws_size, hipStream_t stream)
{
  (void)n_in; (void)out_size; (void)ws_size;

  const float* x       = (const float*)d_in[0];
  const float* c1_Wb   = (const float*)d_in[1];
  const float* c1_bb   = (const float*)d_in[2];
  const float* c1_Wu   = (const float*)d_in[3];
  const float* c1_bu   = (const float*)d_in[4];
  const float* c1_Wb2b = (const float*)d_in[5];
  const float* c1_bb2b = (const float*)d_in[6];
  // d_in[7]=c1_Wu2b, d_in[8]=c1_bu2b (unused: layer-1 pos edges are all p==0)
  const float* c1_Wu2u = (const float*)d_in[9];
  const float* c1_bu2u = (const float*)d_in[10];
  // d_in[11]=c1_Wb2u, d_in[12]=c1_bb2u (unused: layer-1 neg edges all p==0)
  const float* c2_Wb   = (const float*)d_in[13];
  const float* c2_bb   = (const float*)d_in[14];
  const float* c2_Wu   = (const float*)d_in[15];
  const float* c2_bu   = (const float*)d_in[16];
  const float* c2_Wb2b = (const float*)d_in[17];
  const float* c2_bb2b = (const float*)d_in[18];
  const float* c2_Wu2b = (const float*)d_in[19];
  const float* c2_bu2b = (const float*)d_in[20];
  const float* c2_Wu2u = (const float*)d_in[21];
  const float* c2_bu2u = (const float*)d_in[22];
  const float* c2_Wb2u = (const float*)d_in[23];
  const float* c2_bb2u = (const float*)d_in[24];
  const float* Wout    = (const float*)d_in[25];
  const float* bout    = (const float*)d_in[26];
  const int*   pos_e   = (const int*)d_in[27];
  const int*   neg_e   = (const int*)d_in[28];
  const float* pos_w   = (const float*)d_in[29];
  const float* neg_w   = (const float*)d_in[30];

  const int N  = in_sizes[0] / 64;     // 50000 (divisible by 16)
  const int Ep = in_sizes[27] / 2;     // 1.6M
  const int En = in_sizes[28] / 2;     // 0.8M
  const int* ps = pos_e;      const int* pd = pos_e + Ep;
  const int* ns = neg_e;      const int* nd = neg_e + En;

  float* wsf  = (float*)d_ws;
  float* bufE = wsf;                    // [eb|eu], later out1 = zb@[Wb|Wu]
  float* bufA = bufE + (size_t)N * 64;  // acc1 -> z, later acc2 -> z2
  float* buf2 = bufA + (size_t)N * 64;  // out2 = zu@[Wb|Wu]
  float* sbuf = buf2 + (size_t)N * 64;  // softmax denominators (N)
  float* escr = sbuf + N;               // per-edge exp scores (Ep+N+En)

  const dim3 gB(N / 16), bB(128);       // GEMM: wave per 16x16 tile
  const dim3 gE(4096),   bE(256);       // edge kernels: grid-stride waves
  const dim3 gF(2048),   bF(256);

  // ---- Layer 1: e1 = [x@c1_Wb+bb | x@c1_Wu+bu] -------------------------
  gemm16_wmma<<<gB, bB, 0, stream>>>(x, 64, 0, 64, c1_Wb, 32, c1_Wu, 32,
                                     c1_bb, c1_bu, bufE, 64, 0);
  fill_f32<<<gF, bF, 0, stream>>>(bufA, N * 64, 0.f);

  // ob: pos edges + N self-loops, all p==0, x=eb, W=c1_Wb2b
  fill_f32<<<gF, bF, 0, stream>>>(sbuf, N, 0.f);
  edge_pass1<<<gE, bE, 0, stream>>>(ps, pd, Ep, N, nullptr, nullptr, 0,
                                    bufE, bufE, 64, c1_Wb2b, c1_bb2b,
                                    c1_Wb2b, c1_bb2b, escr, sbuf);
  edge_pass2<<<gE, bE, 0, stream>>>(ps, pd, pos_w, Ep, N,
                                    nullptr, nullptr, nullptr, 0,
                                    bufE, bufE, 64, escr, sbuf, bufA, 64, 0);
  // ou: neg edges, all p==0, x=eu, W=c1_Wu2u
  fill_f32<<<gF, bF, 0, stream>>>(sbuf, N, 0.f);
  edge_pass1<<<gE, bE, 0, stream>>>(ns, nd, En, 0, nullptr, nullptr, 0,
                                    bufE + 32, bufE + 32, 64, c1_Wu2u, c1_bu2u,
                                    c1_Wu2u, c1_bu2u, escr, sbuf);
  edge_pass2<<<gE, bE, 0, stream>>>(ns, nd, neg_w, En, 0,
                                    nullptr, nullptr, nullptr, 0,
                                    bufE + 32, bufE + 32, 64, escr, sbuf,
                                    bufA, 64, 32);
  // z = tanh(acc1)
  tanh_ip<<<gF, bF, 0, stream>>>(bufA, N * 64);

  // ---- Layer 2 node transforms -----------------------------------------
  // out1 = zb@[c2_Wb|c2_Wu]+[bb|bu]  (cols 0-31: zb@Wb, 32-63: zb@Wu)
  gemm16_wmma<<<gB, bB, 0, stream>>>(bufA, 64, 0, 32, c2_Wb, 32, c2_Wu, 32,
                                     c2_bb, c2_bu, bufE, 64, 0);
  // out2 = zu@[c2_Wb|c2_Wu]+[bb|bu]
  gemm16_wmma<<<gB, bB, 0, stream>>>(bufA, 64, 32, 32, c2_Wb, 32, c2_Wu, 32,
                                     c2_bb, c2_bu, buf2, 64, 0);
  fill_f32<<<gF, bF, 0, stream>>>(bufA, N * 64, 0.f);   // acc2

  // ob2: p0 -> x1=zb@Wb (out1[:, :32]), W0=c2_Wb2b ;
  //      p1 -> x2=zu@Wb (out2[:, :32]), W1=c2_Wu2b
  fill_f32<<<gF, bF, 0, stream>>>(sbuf, N, 0.f);
  edge_pass1<<<gE, bE, 0, stream>>>(ps, pd, Ep, N, ns, nd, En,
                                    bufE, buf2, 64, c2_Wb2b, c2_bb2b,
                                    c2_Wu2b, c2_bu2b, escr, sbuf);
  edge_pass2<<<gE, bE, 0, stream>>>(ps, pd, pos_w, Ep, N, ns, nd, neg_w, En,
                                    bufE, buf2, 64, escr, sbuf, bufA, 64, 0);
  // ou2: p0 -> x1=zu@Wu (out2[:,32:]), W0=c2_Wu2u ;
  //      p1 -> x2=zb@Wu (out1[:,32:]), W1=c2_Wb2u
  fill_f32<<<gF, bF, 0, stream>>>(sbuf, N, 0.f);
  edge_pass1<<<gE, bE, 0, stream>>>(ps, pd, Ep, N, ns, nd, En,
                                    buf2 + 32, bufE + 32, 64, c2_Wu2u, c2_bu2u,
                                    c2_Wb2u, c2_bb2u, escr, sbuf);
  edge_pass2<<<gE, bE, 0, stream>>>(ps, pd, pos_w, Ep, N, ns, nd, neg_w, En,
                                    buf2 + 32, bufE + 32, 64, escr, sbuf,
                                    bufA, 64, 32);
  // z2 = tanh(acc2)
  tanh_ip<<<gF, bF, 0, stream>>>(bufA, N * 64);

  // ---- Output: tanh(z2 @ Wout + bout) ----------------------------------
  gemm16_wmma<<<gB, bB, 0, stream>>>(bufA, 64, 0, 64, Wout, 64, Wout + 32, 64,
                                     bout, bout + 32, (float*)d_out, 64, 1);
}